// SarathiTransformerBlock_29403346108605
// MI455X (gfx1250) — compile-verified
//
#include <hip/hip_runtime.h>
#include <hip/hip_bf16.h>
#include <math.h>

// ---------------------------------------------------------------------------
// SARATHI transformer block for MI455X (gfx1250), fp32 WMMA (16x16x4_f32).
// Memory-bound workload (~3.4 GB traffic => ~150us floor @ 23.3 TB/s), so we
// keep full fp32 precision; matrix ops still go through v_wmma.
// ---------------------------------------------------------------------------

typedef float v2f __attribute__((ext_vector_type(2)));
typedef float v8f __attribute__((ext_vector_type(8)));

__device__ __forceinline__ v8f wmma4(v2f a, v2f b, v8f c) {
#if defined(__HIP_DEVICE_COMPILE__)
  // D = A(16x4 f32) x B(4x16 f32) + C(16x16 f32)
  return __builtin_amdgcn_wmma_f32_16x16x4_f32(
      /*neg_a=*/false, a, /*neg_b=*/false, b,
      /*c_mod=*/(short)0, c, /*reuse_a=*/false, /*reuse_b=*/false);
#else
  (void)a; (void)b;
  return c;   // host pass: never executed, only parsed
#endif
}

// Problem constants
#define DMODEL 2048
#define NHEAD  16
#define DHEAD  128
#define FFDIM  8192
#define SCHUNK 512
#define SPAST  2048
#define BDEC   32
#define KVLEN  2048
#define MTOT   544            // 512 prefill + 32 decode rows
#define QKVW   (3*DMODEL)     // 6144
#define ATTN_SCALE 0.08838834764831845f   // 1/sqrt(128)

// d_out layout (floats, concatenated in reference return order)
#define OFF_FINAL_P 0
#define OFF_NPK     1048576
#define OFF_NPV     6291456
#define OFF_FINAL_D 11534336
#define OFF_NDK     11599872
#define OFF_NDV     145883136

// ---------------------------------------------------------------------------
// LayerNorm: one block per row (544 rows); rows<512 from srcA, else srcB.
// ---------------------------------------------------------------------------
__global__ void ln_kernel(const float* __restrict__ srcA,
                          const float* __restrict__ srcB,
                          const float* __restrict__ g,
                          const float* __restrict__ bb,
                          float* __restrict__ dst) {
  __shared__ float red[256];
  const int row = blockIdx.x;
  const int tid = threadIdx.x;
  const float* x = (row < SCHUNK) ? (srcA + (size_t)row * DMODEL)
                                  : (srcB + (size_t)(row - SCHUNK) * DMODEL);
  float xr[8];
  float s = 0.f;
#pragma unroll
  for (int i = 0; i < 8; ++i) { xr[i] = x[tid + i * 256]; s += xr[i]; }
  red[tid] = s; __syncthreads();
  for (int off = 128; off > 0; off >>= 1) {
    if (tid < off) red[tid] += red[tid + off];
    __syncthreads();
  }
  const float mu = red[0] * (1.0f / DMODEL);
  __syncthreads();
  float vs = 0.f;
#pragma unroll
  for (int i = 0; i < 8; ++i) { float d = xr[i] - mu; vs += d * d; }
  red[tid] = vs; __syncthreads();
  for (int off = 128; off > 0; off >>= 1) {
    if (tid < off) red[tid] += red[tid + off];
    __syncthreads();
  }
  const float inv = rsqrtf(red[0] * (1.0f / DMODEL) + 1e-5f);
  float* out = dst + (size_t)row * DMODEL;
#pragma unroll
  for (int i = 0; i < 8; ++i) {
    const int c = tid + i * 256;
    out[c] = (xr[i] - mu) * inv * g[c] + bb[c];
  }
}

// ---------------------------------------------------------------------------
// Generic fp32 WMMA GEMM: C[544,N] = A[544,K] @ B[K,N] + bias (+relu)(+resid)
// Grid: (N/128, 34). Block: 256 threads = 8 waves; each wave owns a 16x16 tile.
// Rows >= 512 route to outB/resB (decode-batch split).
// ---------------------------------------------------------------------------
__global__ void gemm_wmma(const float* __restrict__ A, int lda,
                          const float* __restrict__ B, int ldb, int K,
                          const float* __restrict__ bias,
                          const float* __restrict__ resA,
                          const float* __restrict__ resB,
                          float* __restrict__ outA,
                          float* __restrict__ outB,
                          int ldc, int relu) {
  const int wid  = threadIdx.x >> 5;
  const int lane = threadIdx.x & 31;
  const int half = lane >> 4;
  const int l16  = lane & 15;

  const int row0 = blockIdx.y * 16;
  const int n    = blockIdx.x * 128 + wid * 16 + l16;

  const float* Arow = A + (size_t)(row0 + l16) * lda;   // A-frag row = l16
  v8f acc = {};
#pragma unroll 8
  for (int k0 = 0; k0 < K; k0 += 4) {
    v2f a = *(const v2f*)(Arow + k0 + 2 * half);        // K {0,1} | {2,3}
    v2f b;
    b.x = B[(size_t)(k0 + 2 * half)     * ldb + n];     // B rows K={0,2}
    b.y = B[(size_t)(k0 + 2 * half + 1) * ldb + n];     // B rows K={1,3}
    acc = wmma4(a, b, acc);
  }

  const float bs = bias[n];
#pragma unroll
  for (int c = 0; c < 8; ++c) {
    const int row = row0 + c + 8 * half;                // C layout: M = c (+8)
    float v = acc[c] + bs;
    if (relu) v = fmaxf(v, 0.f);
    if (row < SCHUNK) {
      if (resA) v += resA[(size_t)row * ldc + n];
      outA[(size_t)row * ldc + n] = v;
    } else {
      const int r2 = row - SCHUNK;
      if (resB) v += resB[(size_t)r2 * ldc + n];
      outB[(size_t)r2 * ldc + n] = v;
    }
  }
}

// ---------------------------------------------------------------------------
// Plane copies (b128) for KV concatenation into d_out.
// ---------------------------------------------------------------------------
__global__ void copy_planes_f4(const float4* __restrict__ src,
                               float4* __restrict__ dst,
                               int plane_f4, long src_stride, long dst_stride) {
  const long p = blockIdx.y;
  const float4* s = src + p * src_stride;
  float4* d = dst + p * dst_stride;
  for (int i = blockIdx.x * blockDim.x + threadIdx.x; i < plane_f4;
       i += gridDim.x * blockDim.x)
    d[i] = s[i];
}

// New prefill K/V rows (chunk tokens) -> npk/npv rows 2048..2559
__global__ void prefill_newkv(const float* __restrict__ qkv,
                              float* __restrict__ npk, float* __restrict__ npv) {
  const int idx = blockIdx.x * 256 + threadIdx.x;   // 16*512*32 = 262144 f4
  if (idx >= 16 * 512 * 32) return;
  const int d4 = idx & 31;
  const int r  = (idx >> 5) & 511;
  const int h  = idx >> 14;
  const float4* sk = (const float4*)(qkv + (size_t)r * QKVW + DMODEL   + h * DHEAD) + d4;
  const float4* sv = (const float4*)(qkv + (size_t)r * QKVW + 2*DMODEL + h * DHEAD) + d4;
  float4* dk = (float4*)(npk + ((size_t)h * 2560 + 2048 + r) * DHEAD) + d4;
  float4* dv = (float4*)(npv + ((size_t)h * 2560 + 2048 + r) * DHEAD) + d4;
  *dk = *sk; *dv = *sv;
}

// New decode K/V rows (1 token per request) -> ndk/ndv row 2048
__global__ void decode_newkv(const float* __restrict__ qkv,
                             float* __restrict__ ndk, float* __restrict__ ndv) {
  const int idx = blockIdx.x * 256 + threadIdx.x;   // 32*16*32 = 16384 f4
  if (idx >= 32 * 16 * 32) return;
  const int d4 = idx & 31;
  const int h  = (idx >> 5) & 15;
  const int b  = idx >> 9;
  const float4* sk = (const float4*)(qkv + (size_t)(SCHUNK + b) * QKVW + DMODEL   + h * DHEAD) + d4;
  const float4* sv = (const float4*)(qkv + (size_t)(SCHUNK + b) * QKVW + 2*DMODEL + h * DHEAD) + d4;
  float4* dk = (float4*)(ndk + ((size_t)(b * 16 + h) * 2049 + 2048) * DHEAD) + d4;
  float4* dv = (float4*)(ndv + ((size_t)(b * 16 + h) * 2049 + 2048) * DHEAD) + d4;
  *dk = *sk; *dv = *sv;
}

// ---------------------------------------------------------------------------
// Prefill flash attention (causal). One wave = one (head, 16-row q-tile).
// QK^T and P*V on the f32 WMMA pipe; P re-laid via per-wave LDS scratch.
// ---------------------------------------------------------------------------
__global__ void prefill_attn(const float* __restrict__ qkv,
                             const float* __restrict__ npk,
                             const float* __restrict__ npv,
                             float* __restrict__ attn) {
  __shared__ float pbuf[8 * 256];                  // 16x16 P tile per wave
  const int wid  = threadIdx.x >> 5;
  const int lane = threadIdx.x & 31;
  const int half = lane >> 4;
  const int l16  = lane & 15;
  const int unit = blockIdx.x * 8 + wid;           // 0..511
  const int head  = unit >> 5;
  const int qtile = unit & 31;

  const float* K = npk + (size_t)head * 2560 * DHEAD;
  const float* V = npv + (size_t)head * 2560 * DHEAD;

  // Preload Q A-fragments for all 32 K-steps (row m = l16, dims 0..127)
  v2f aq[32];
  const float* Qr = qkv + (size_t)(qtile * 16 + l16) * QKVW + head * DHEAD;
#pragma unroll
  for (int kk = 0; kk < 32; ++kk)
    aq[kk] = *(const v2f*)(Qr + kk * 4 + 2 * half);

  v8f o[8];
  float m[8], l[8];
#pragma unroll
  for (int t = 0; t < 8; ++t) o[t] = (v8f){};
#pragma unroll
  for (int c = 0; c < 8; ++c) { m[c] = -1.0e30f; l[c] = 0.f; }

  const int kend = 2064 + qtile * 16;              // causal bound, <= 2560
  for (int kb = 0; kb < kend; kb += 16) {
    // ---- S = Q * K^T  (16x16 tile, K-dim = 128 => 32 wmma steps)
    const float* Krow = K + (size_t)(kb + l16) * DHEAD;
    v8f s = {};
#pragma unroll
    for (int kk = 0; kk < 32; ++kk) {
      v2f b = *(const v2f*)(Krow + kk * 4 + 2 * half);
      s = wmma4(aq[kk], b, s);
    }
    // ---- online softmax (per row c + 8*half; cols live in half-wave lanes)
    float scv[8];
#pragma unroll
    for (int c = 0; c < 8; ++c) {
      const int row = c + 8 * half;
      float sv = s[c] * ATTN_SCALE;
      if (kb + l16 > 2048 + qtile * 16 + row) sv = -1.0e30f;   // causal mask
      float rm = sv;
      rm = fmaxf(rm, __shfl_xor(rm, 1, 32));
      rm = fmaxf(rm, __shfl_xor(rm, 2, 32));
      rm = fmaxf(rm, __shfl_xor(rm, 4, 32));
      rm = fmaxf(rm, __shfl_xor(rm, 8, 32));
      const float mn = fmaxf(m[c], rm);
      const float sc = __expf(m[c] - mn);
      const float p  = __expf(sv - mn);
      float ps = p;
      ps += __shfl_xor(ps, 1, 32);
      ps += __shfl_xor(ps, 2, 32);
      ps += __shfl_xor(ps, 4, 32);
      ps += __shfl_xor(ps, 8, 32);
      l[c] = l[c] * sc + ps;
      m[c] = mn;
      scv[c] = sc;
      pbuf[wid * 256 + row * 16 + l16] = p;        // C-layout -> LDS
    }
#pragma unroll
    for (int t = 0; t < 8; ++t)
#pragma unroll
      for (int c = 0; c < 8; ++c) o[t][c] *= scv[c];
    // ---- O += P * V   (A-frag from LDS, K-dim = 16 keys => 4 wmma steps)
#pragma unroll
    for (int kk = 0; kk < 4; ++kk) {
      v2f a = *(const v2f*)(&pbuf[wid * 256 + l16 * 16 + kk * 4 + 2 * half]);
#pragma unroll
      for (int t = 0; t < 8; ++t) {
        const float* vp = V + (size_t)(kb + kk * 4 + 2 * half) * DHEAD + t * 16 + l16;
        v2f b; b.x = vp[0]; b.y = vp[DHEAD];
        o[t] = wmma4(a, b, o[t]);
      }
    }
  }
  // ---- write O / l
#pragma unroll
  for (int c = 0; c < 8; ++c) {
    const float inv = 1.0f / l[c];
    const int row = qtile * 16 + c + 8 * half;
#pragma unroll
    for (int t = 0; t < 8; ++t)
      attn[(size_t)row * DMODEL + head * DHEAD + t * 16 + l16] = o[t][c] * inv;
  }
}

// ---------------------------------------------------------------------------
// Decode attention: 1 query x 2049 keys per (b,h). GEMV-style, memory-bound.
// One block per (b,h); 8 waves split keys; online softmax; LDS merge.
// ---------------------------------------------------------------------------
__global__ void decode_attn(const float* __restrict__ qkv,
                            const float* __restrict__ ndk,
                            const float* __restrict__ ndv,
                            float* __restrict__ attn) {
  __shared__ float sm[8], sl[8], sacc[8][DHEAD];
  const int b = blockIdx.x >> 4;
  const int h = blockIdx.x & 15;
  const int wid  = threadIdx.x >> 5;
  const int lane = threadIdx.x & 31;
  const float* K = ndk + (size_t)(b * 16 + h) * 2049 * DHEAD;
  const float* V = ndv + (size_t)(b * 16 + h) * 2049 * DHEAD;
  const float4 q = *(const float4*)(qkv + (size_t)(SCHUNK + b) * QKVW + h * DHEAD + lane * 4);

  float m = -1.0e30f, l = 0.f;
  float4 acc = make_float4(0.f, 0.f, 0.f, 0.f);
  for (int j = wid; j < 2049; j += 8) {
    const float4 kv = *(const float4*)(K + (size_t)j * DHEAD + lane * 4);
    float s = q.x * kv.x + q.y * kv.y + q.z * kv.z + q.w * kv.w;
    s += __shfl_xor(s, 1, 32);
    s += __shfl_xor(s, 2, 32);
    s += __shfl_xor(s, 4, 32);
    s += __shfl_xor(s, 8, 32);
    s += __shfl_xor(s, 16, 32);
    s *= ATTN_SCALE;
    const float mn = fmaxf(m, s);
    const float sc = __expf(m - mn);
    const float p  = __expf(s - mn);
    m = mn; l = l * sc + p;
    const float4 vv = *(const float4*)(V + (size_t)j * DHEAD + lane * 4);
    acc.x = acc.x * sc + p * vv.x;
    acc.y = acc.y * sc + p * vv.y;
    acc.z = acc.z * sc + p * vv.z;
    acc.w = acc.w * sc + p * vv.w;
  }
  if (lane == 0) { sm[wid] = m; sl[wid] = l; }
  *(float4*)(&sacc[wid][lane * 4]) = acc;
  __syncthreads();
  if (threadIdx.x < DHEAD) {
    float M = sm[0];
#pragma unroll
    for (int w = 1; w < 8; ++w) M = fmaxf(M, sm[w]);
    float L = 0.f, O = 0.f;
#pragma unroll
    for (int w = 0; w < 8; ++w) {
      const float e = __expf(sm[w] - M);
      L += sl[w] * e;
      O += sacc[w][threadIdx.x] * e;
    }
    attn[(size_t)(SCHUNK + b) * DMODEL + h * DHEAD + threadIdx.x] = O / L;
  }
}

// ---------------------------------------------------------------------------
extern "C" void kernel_launch(void* const* d_in, const int* in_sizes, int n_in,
                              void* d_out, int out_size, void* d_ws, size_t ws_size,
                              hipStream_t stream) {
  (void)in_sizes; (void)n_in; (void)out_size; (void)ws_size;
  const float* prefill_chunk = (const float*)d_in[0];
  const float* prefill_k     = (const float*)d_in[1];
  const float* prefill_v     = (const float*)d_in[2];
  const float* decode_tokens = (const float*)d_in[3];
  const float* decode_k      = (const float*)d_in[4];
  const float* decode_v      = (const float*)d_in[5];
  const float* qkv_w = (const float*)d_in[6];
  const float* qkv_b = (const float*)d_in[7];
  const float* out_w = (const float*)d_in[8];
  const float* out_b = (const float*)d_in[9];
  const float* w1    = (const float*)d_in[10];
  const float* b1    = (const float*)d_in[11];
  const float* w2    = (const float*)d_in[12];
  const float* b2    = (const float*)d_in[13];
  const float* ln1_g = (const float*)d_in[14];
  const float* ln1_b = (const float*)d_in[15];
  const float* ln2_g = (const float*)d_in[16];
  const float* ln2_b = (const float*)d_in[17];

  float* out = (float*)d_out;
  float* npk = out + OFF_NPK;
  float* npv = out + OFF_NPV;
  float* ndk = out + OFF_NDK;
  float* ndv = out + OFF_NDV;

  // Workspace layout (floats). buf_big is reused: qkv -> ffn hidden.
  float* ws    = (float*)d_ws;
  float* xln   = ws;                               // 544*2048 (also LN2 out)
  float* qkv   = ws + 1114112;                     // 544*6144 / 544*8192
  float* hbuf  = qkv;
  float* attn  = ws + 5570560;                     // 544*2048
  float* fused = ws + 6684672;                     // 544*2048

  // 1) LN1
  ln_kernel<<<MTOT, 256, 0, stream>>>(prefill_chunk, decode_tokens, ln1_g, ln1_b, xln);
  // 2) QKV projection
  gemm_wmma<<<dim3(QKVW / 128, MTOT / 16), 256, 0, stream>>>(
      xln, DMODEL, qkv_w, QKVW, DMODEL, qkv_b, nullptr, nullptr,
      qkv, qkv + (size_t)SCHUNK * QKVW, QKVW, 0);
  // 3) KV concatenation (b128 plane copies) + new-token rows
  copy_planes_f4<<<dim3(64, 16), 256, 0, stream>>>(
      (const float4*)prefill_k, (float4*)npk, 65536, 65536, 81920);
  copy_planes_f4<<<dim3(64, 16), 256, 0, stream>>>(
      (const float4*)prefill_v, (float4*)npv, 65536, 65536, 81920);
  copy_planes_f4<<<dim3(64, 512), 256, 0, stream>>>(
      (const float4*)decode_k, (float4*)ndk, 65536, 65536, 65568);
  copy_planes_f4<<<dim3(64, 512), 256, 0, stream>>>(
      (const float4*)decode_v, (float4*)ndv, 65536, 65536, 65568);
  prefill_newkv<<<1024, 256, 0, stream>>>(qkv, npk, npv);
  decode_newkv<<<64, 256, 0, stream>>>(qkv, ndk, ndv);
  // 4) Attention (reads concatenated KV directly from d_out)
  prefill_attn<<<64, 256, 0, stream>>>(qkv, npk, npv, attn);
  decode_attn<<<512, 256, 0, stream>>>(qkv, ndk, ndv, attn);
  // 5) Output projection + residual
  gemm_wmma<<<dim3(DMODEL / 128, MTOT / 16), 256, 0, stream>>>(
      attn, DMODEL, out_w, DMODEL, DMODEL, out_b, prefill_chunk, decode_tokens,
      fused, fused + (size_t)SCHUNK * DMODEL, DMODEL, 0);
  // 6) LN2
  ln_kernel<<<MTOT, 256, 0, stream>>>(fused, fused + (size_t)SCHUNK * DMODEL,
                                      ln2_g, ln2_b, xln);
  // 7) FFN up + ReLU
  gemm_wmma<<<dim3(FFDIM / 128, MTOT / 16), 256, 0, stream>>>(
      xln, DMODEL, w1, FFDIM, DMODEL, b1, nullptr, nullptr,
      hbuf, hbuf + (size_t)SCHUNK * FFDIM, FFDIM, 1);
  // 8) FFN down + residual, split-write final_p / final_d
  gemm_wmma<<<dim3(DMODEL / 128, MTOT / 16), 256, 0, stream>>>(
      hbuf, FFDIM, w2, DMODEL, FFDIM, b2, fused, fused + (size_t)SCHUNK * DMODEL,
      out + OFF_FINAL_P, out + OFF_FINAL_D, DMODEL, 0);
}